// SSM_8014408974608
// MI455X (gfx1250) — compile-verified
//
#include <hip/hip_runtime.h>
#include <stdint.h>

// Problem dims (from reference)
#define BATCH 4
#define SEQ   2048
#define DDIM  768
#define NDIM  16

// Chunked-scan config
#define CH    8                       // sequence chunks (parallelism for BW)
#define CLEN  (SEQ / CH)              // 256 steps per chunk
#define TS    16                      // timesteps per TDM stage
#define NSTG  (CLEN / TS)             // 16 stages per chunk
#define DTILE 16                      // d-values per block
#define TILE_ELEMS (DTILE * NDIM)     // 256 contiguous f32 per timestep tile
#define ROWSTRIDE  (DDIM * NDIM)      // 12288 elems between timesteps
#define BDN   (BATCH * DDIM * NDIM)   // 49152 recurrence channels

typedef unsigned int u32;
typedef __attribute__((ext_vector_type(4))) u32 u32x4_t;
typedef __attribute__((ext_vector_type(8))) int i32x8_t;
typedef __attribute__((ext_vector_type(4))) int i32x4_t;

// ---------------------------------------------------------------------------
// Tensor Data Mover: 2D tile load (rows x TILE_ELEMS f32, row stride
// ROWSTRIDE elems) from global into contiguous LDS. D# packed per
// CDNA5 ISA. Issued per-wave (EXEC ignored),
// tracked with TENSORcnt. 6-arg builtin (clang-23 / therock-10.0 flavor).
// ---------------------------------------------------------------------------
__device__ __forceinline__ void tdm_load_tile_2d(u32 lds_off, const void* gptr,
                                                 u32 rows) {
  unsigned long long ga = (unsigned long long)(uintptr_t)gptr;
  u32x4_t g0;
  g0.x = 1u;                                              // count=1 (valid), user mode
  g0.y = lds_off;                                         // lds_addr (bytes)
  g0.z = (u32)(ga & 0xFFFFFFFFull);                       // global_addr[31:0]
  g0.w = (u32)((ga >> 32) & 0x1FFFFFFull) | (2u << 30);   // global_addr[56:32] | type=2
  i32x8_t g1;
  g1[0] = (int)(2u << 16);                    // wg_mask=0, data_size=2 (4 bytes)
  g1[1] = (int)((u32)TILE_ELEMS << 16);       // tensor_dim0[15:0] (bits 79:48 lo)
  g1[2] = (int)(rows << 16);                  // tensor_dim0[31:16]=0 | tensor_dim1[15:0]
  g1[3] = (int)((u32)TILE_ELEMS << 16);       // tensor_dim1[31:16]=0 | tile_dim0
  g1[4] = (int)rows;                          // tile_dim1=rows, tile_dim2=0
  g1[5] = (int)ROWSTRIDE;                     // tensor_dim0_stride[31:0]
  g1[6] = 0;                                  // stride0[47:32] | stride1[15:0]
  g1[7] = 0;                                  // stride1[47:16]
  i32x4_t z4 = {0, 0, 0, 0};                  // 2D tensor: groups 2/3 unused
  i32x8_t z8 = {0, 0, 0, 0, 0, 0, 0, 0};
  __builtin_amdgcn_tensor_load_to_lds(g0, g1, z4, z4, z8, 0);
}

__device__ __forceinline__ u32 lds_offset_of(const void* p) {
  // Generic pointer to LDS: addr[31:0] is the workgroup-relative LDS offset.
  return (u32)(uintptr_t)p;
}

// ---------------------------------------------------------------------------
// Kernel 1: per-chunk local scan (h0 = 0). One block per (b, chunk, d-tile).
// 256 threads: lane = dl*16 + n  (matches contiguous [d][n] tile layout).
// Streams A,B tiles via TDM double-buffering; writes y_local, chunk P & h_end.
// LDS: 2 bufs x 2 arrays x 16KB = 64KB -> 5 blocks (40 waves) per WGP.
// ---------------------------------------------------------------------------
__global__ void __launch_bounds__(256) ssm_local_scan(
    const float* __restrict__ A, const float* __restrict__ B,
    const float* __restrict__ C, const float* __restrict__ x,
    float* __restrict__ y, float* __restrict__ Pc, float* __restrict__ Hc) {
  __shared__ float ldsA[2][TS * TILE_ELEMS];
  __shared__ float ldsB[2][TS * TILE_ELEMS];

  const int tid = threadIdx.x;
  const int n   = tid & (NDIM - 1);
  const int dl  = tid >> 4;
  const int dt  = blockIdx.x;
  const int ck  = blockIdx.y;
  const int b   = blockIdx.z;
  const int d0  = dt * DTILE;
  const int t0  = ck * CLEN;
  const int d   = d0 + dl;

  const unsigned long long baseElem =
      (((unsigned long long)b * SEQ + t0) * DDIM + d0) * NDIM;
  const char* gA = (const char*)A + baseElem * 4ull;
  const char* gB = (const char*)B + baseElem * 4ull;
  const unsigned long long stageBytes = (unsigned long long)TS * ROWSTRIDE * 4ull;

  if (tid < 32) {  // wave 0 drives the TDM pipeline
    tdm_load_tile_2d(lds_offset_of(&ldsA[0][0]), gA, TS);
    tdm_load_tile_2d(lds_offset_of(&ldsB[0][0]), gB, TS);
  }

  float h = 0.0f;   // recurrence state for this (b,d,n)
  float p = 1.0f;   // running product of A over the chunk

  for (int s = 0; s < NSTG; ++s) {
    const int cur = s & 1;
    if (tid < 32) {
      if (s + 1 < NSTG) {
        const char* gAn = gA + (unsigned long long)(s + 1) * stageBytes;
        const char* gBn = gB + (unsigned long long)(s + 1) * stageBytes;
        tdm_load_tile_2d(lds_offset_of(&ldsA[cur ^ 1][0]), gAn, TS);
        tdm_load_tile_2d(lds_offset_of(&ldsB[cur ^ 1][0]), gBn, TS);
        __builtin_amdgcn_s_wait_tensorcnt(2);  // current stage's 2 TDMs done
      } else {
        __builtin_amdgcn_s_wait_tensorcnt(0);
      }
    }
    __syncthreads();  // stage data visible to all waves

#pragma unroll
    for (int tt = 0; tt < TS; ++tt) {
      const int t = t0 + s * TS + tt;
      const float a  = ldsA[cur][tt * TILE_ELEMS + tid];
      const float bv = ldsB[cur][tt * TILE_ELEMS + tid];
      const float xv = x[((unsigned long long)b * SEQ + t) * DDIM + d];
      const float cv = C[((unsigned long long)b * SEQ + t) * NDIM + n];
      h = fmaf(a, h, bv * xv);
      p *= a;
      float v = cv * h;                 // contract over n (16 lanes)
      v += __shfl_xor(v, 1, 32);
      v += __shfl_xor(v, 2, 32);
      v += __shfl_xor(v, 4, 32);
      v += __shfl_xor(v, 8, 32);
      if (n == 0) y[((unsigned long long)b * SEQ + t) * DDIM + d] = v;
    }
    __syncthreads();  // protect buffer before TDM reuses it
  }

  const unsigned long long bdn = ((unsigned long long)b * DDIM + d) * NDIM + n;
  Pc[(unsigned long long)ck * BDN + bdn] = p;
  Hc[(unsigned long long)ck * BDN + bdn] = h;
}

// ---------------------------------------------------------------------------
// Kernel 2: serial combine over the 8 chunks per channel -> exclusive carries.
// ---------------------------------------------------------------------------
__global__ void __launch_bounds__(256) ssm_chunk_carry(
    const float* __restrict__ Pc, const float* __restrict__ Hc,
    float* __restrict__ Carry) {
  const int i = blockIdx.x * blockDim.x + threadIdx.x;  // 0..BDN-1
  float H = 0.0f;
#pragma unroll
  for (int k = 0; k < CH; ++k) {
    Carry[k * BDN + i] = H;                 // state entering chunk k
    H = Pc[k * BDN + i] * H + Hc[k * BDN + i];
  }
}

// ---------------------------------------------------------------------------
// Kernel 3: fixup for chunks 1..CH-1. The correction is the homogeneous
// recurrence z_t = A_t * z_{t-1} seeded with the carry; y += C . z.
// Only A is re-streamed (via TDM). LDS: 32KB/block.
// ---------------------------------------------------------------------------
__global__ void __launch_bounds__(256) ssm_fixup(
    const float* __restrict__ A, const float* __restrict__ C,
    const float* __restrict__ Carry, float* __restrict__ y) {
  __shared__ float ldsA[2][TS * TILE_ELEMS];

  const int tid = threadIdx.x;
  const int n   = tid & (NDIM - 1);
  const int dl  = tid >> 4;
  const int dt  = blockIdx.x;
  const int ck  = blockIdx.y + 1;  // chunk 0 has zero carry-in
  const int b   = blockIdx.z;
  const int d0  = dt * DTILE;
  const int t0  = ck * CLEN;
  const int d   = d0 + dl;

  const unsigned long long baseElem =
      (((unsigned long long)b * SEQ + t0) * DDIM + d0) * NDIM;
  const char* gA = (const char*)A + baseElem * 4ull;
  const unsigned long long stageBytes = (unsigned long long)TS * ROWSTRIDE * 4ull;

  if (tid < 32) tdm_load_tile_2d(lds_offset_of(&ldsA[0][0]), gA, TS);

  const unsigned long long bdn = ((unsigned long long)b * DDIM + d) * NDIM + n;
  float h = Carry[(unsigned long long)ck * BDN + bdn];

  for (int s = 0; s < NSTG; ++s) {
    const int cur = s & 1;
    if (tid < 32) {
      if (s + 1 < NSTG) {
        const char* gAn = gA + (unsigned long long)(s + 1) * stageBytes;
        tdm_load_tile_2d(lds_offset_of(&ldsA[cur ^ 1][0]), gAn, TS);
        __builtin_amdgcn_s_wait_tensorcnt(1);
      } else {
        __builtin_amdgcn_s_wait_tensorcnt(0);
      }
    }
    __syncthreads();

#pragma unroll
    for (int tt = 0; tt < TS; ++tt) {
      const int t = t0 + s * TS + tt;
      const float a  = ldsA[cur][tt * TILE_ELEMS + tid];
      const float cv = C[((unsigned long long)b * SEQ + t) * NDIM + n];
      h *= a;
      float v = cv * h;
      v += __shfl_xor(v, 1, 32);
      v += __shfl_xor(v, 2, 32);
      v += __shfl_xor(v, 4, 32);
      v += __shfl_xor(v, 8, 32);
      if (n == 0) {
        const unsigned long long yi = ((unsigned long long)b * SEQ + t) * DDIM + d;
        y[yi] = y[yi] + v;   // unique (b,t,d) per block: plain RMW is safe
      }
    }
    __syncthreads();
  }
}

// ---------------------------------------------------------------------------
// Host launcher. Inputs (setup_inputs order): A, B, C, x. Output: y [b,s,d].
// ---------------------------------------------------------------------------
extern "C" void kernel_launch(void* const* d_in, const int* in_sizes, int n_in,
                              void* d_out, int out_size, void* d_ws, size_t ws_size,
                              hipStream_t stream) {
  const float* A = (const float*)d_in[0];
  const float* B = (const float*)d_in[1];
  const float* C = (const float*)d_in[2];
  const float* x = (const float*)d_in[3];
  float* y = (float*)d_out;

  float* Pc    = (float*)d_ws;                 // [CH][BDN] chunk A-products
  float* Hc    = Pc + (size_t)CH * BDN;        // [CH][BDN] chunk local end-states
  float* Carry = Hc + (size_t)CH * BDN;        // [CH][BDN] exclusive carries

  dim3 blk(256);
  dim3 g1(DDIM / DTILE, CH, BATCH);            // 48 x 8 x 4 = 1536 blocks
  ssm_local_scan<<<g1, blk, 0, stream>>>(A, B, C, x, y, Pc, Hc);

  dim3 g2(BDN / 256);                          // 192 blocks
  ssm_chunk_carry<<<g2, blk, 0, stream>>>(Pc, Hc, Carry);

  dim3 g3(DDIM / DTILE, CH - 1, BATCH);        // 48 x 7 x 4 = 1344 blocks
  ssm_fixup<<<g3, blk, 0, stream>>>(A, C, Carry, y);
}